// RelativeRole_35029753266256
// MI455X (gfx1250) — compile-verified
//
#include <hip/hip_runtime.h>
#include <hip/hip_bf16.h>
#include <math.h>

// Problem constants (match reference)
#define BS      2
#define QLEN    64
#define KLEN    64
#define NHEADS  8
#define DKV     64
#define INNER   512          // NHEADS * DKV
#define VOCAB   4096
#define MROWS   (BS*QLEN)    // 128 flattened q rows
#define TOKR    (BS*QLEN*NHEADS)  // 1024 rows of [*,64] head-vectors

typedef float v2f __attribute__((ext_vector_type(2)));
typedef float v8f __attribute__((ext_vector_type(8)));

// ---------------------------------------------------------------------------
// Elementwise: q = query + qpos ; k = key + kpos
// ---------------------------------------------------------------------------
__global__ void addpos_kernel(const float* __restrict__ q, const float* __restrict__ qp,
                              const float* __restrict__ k, const float* __restrict__ kp,
                              float* __restrict__ qs, float* __restrict__ ks, int n) {
  int i = blockIdx.x * blockDim.x + threadIdx.x;
  if (i < n) { qs[i] = q[i] + qp[i]; ks[i] = k[i] + kp[i]; }
}

__global__ void zero_kernel(float* __restrict__ p, int n) {
  int i = blockIdx.x * blockDim.x + threadIdx.x;
  if (i < n) p[i] = 0.f;
}

// ---------------------------------------------------------------------------
// WMMA f32 16x16x4 GEMM, C = A(MxK) @ B(KxN), row-major. One 16x16 tile/wave.
// A layout (16x4 f32): lane half splits K: lanes0-15 K=k0,k0+1 ; lanes16-31 K=k0+2,k0+3
// B layout (4x16): VGPR j, lanes0-15 K=k0+j ; lanes16-31 K=k0+2+j, N = lane%16
// C/D: VGPR i -> M = i + 8*half, N = lane%16
// ---------------------------------------------------------------------------
__global__ __launch_bounds__(128) void gemm_nn_wmma(
    const float* __restrict__ A, const float* __restrict__ B, float* __restrict__ C,
    int M, int N, int K, int lda, int ldb, int ldc) {
  int wave = threadIdx.x >> 5, lane = threadIdx.x & 31;
  int half = lane >> 4, r = lane & 15;
  int row0 = blockIdx.y * 16;
  int col0 = blockIdx.x * 64 + wave * 16;
  if (row0 >= M || col0 >= N) return;           // wave-uniform guard, EXEC stays all-1
  v8f acc = {};
  const float* Arow = A + (size_t)(row0 + r) * lda;
  for (int k0 = 0; k0 < K; k0 += 4) {
    v2f a, b;
    a.x = Arow[k0 + 2*half + 0];
    a.y = Arow[k0 + 2*half + 1];
    b.x = B[(size_t)(k0 + 2*half + 0) * ldb + col0 + r];
    b.y = B[(size_t)(k0 + 2*half + 1) * ldb + col0 + r];
    acc = __builtin_amdgcn_wmma_f32_16x16x4_f32(false, a, false, b, (short)0, acc, false, false);
  }
  for (int i = 0; i < 8; ++i)
    C[(size_t)(row0 + i + 8*half) * ldc + col0 + r] = acc[i];
}

// C = A(MxK) @ B^T, with B stored (NxK) row-major. Batched via grid.z with
// two-level offsets: z -> (zo, zi); ptr += zo*SO + zi*SI.
__global__ __launch_bounds__(128) void gemm_nt_wmma(
    const float* __restrict__ A, const float* __restrict__ B, float* __restrict__ C,
    int M, int N, int K, int lda, int ldb, int ldc,
    int zInner, int aSO, int aSI, int bSO, int bSI, int cSO, int cSI) {
  int z = blockIdx.z, zo = z / zInner, zi = z - zo * zInner;
  A += (size_t)zo * aSO + (size_t)zi * aSI;
  B += (size_t)zo * bSO + (size_t)zi * bSI;
  C += (size_t)zo * cSO + (size_t)zi * cSI;
  int wave = threadIdx.x >> 5, lane = threadIdx.x & 31;
  int half = lane >> 4, r = lane & 15;
  int row0 = blockIdx.y * 16;
  int col0 = blockIdx.x * 64 + wave * 16;
  if (row0 >= M || col0 >= N) return;
  v8f acc = {};
  const float* Arow = A + (size_t)(row0 + r) * lda;
  const float* Brow = B + (size_t)(col0 + r) * ldb;
  for (int k0 = 0; k0 < K; k0 += 4) {
    v2f a, b;
    a.x = Arow[k0 + 2*half + 0];
    a.y = Arow[k0 + 2*half + 1];
    b.x = Brow[k0 + 2*half + 0];
    b.y = Brow[k0 + 2*half + 1];
    acc = __builtin_amdgcn_wmma_f32_16x16x4_f32(false, a, false, b, (short)0, acc, false, false);
  }
  for (int i = 0; i < 8; ++i)
    C[(size_t)(row0 + i + 8*half) * ldc + col0 + r] = acc[i];
}

// ---------------------------------------------------------------------------
// d_n[v] = vq_dict[v] * rsqrt(max(||vq_dict[v]||^2, 1e-12))
// ---------------------------------------------------------------------------
__global__ void dictnorm_kernel(const float* __restrict__ dict, float* __restrict__ dn) {
  int v = blockIdx.x * blockDim.x + threadIdx.x;
  if (v >= VOCAB) return;
  float sq = 0.f;
  for (int c = 0; c < DKV; ++c) { float x = dict[v*DKV + c]; sq += x*x; }
  float inv = rsqrtf(fmaxf(sq, 1e-12f));
  for (int c = 0; c < DKV; ++c) dn[v*DKV + c] = dict[v*DKV + c] * inv;
}

// ||qp_head||^2 and ||kp_head||^2 per (b,q,h)/(b,k,h) row (1024 rows each)
__global__ void rownorm2_kernel(const float* __restrict__ qproj, const float* __restrict__ kproj,
                                float* __restrict__ qn2, float* __restrict__ kn2) {
  int m = blockIdx.x * blockDim.x + threadIdx.x;
  if (m >= TOKR) return;
  float sq = 0.f, sk = 0.f;
  for (int c = 0; c < DKV; ++c) {
    float a = qproj[m*DKV + c]; sq += a*a;
    float b = kproj[m*DKV + c]; sk += b*b;
  }
  qn2[m] = sq; kn2[m] = sk;
}

// ---------------------------------------------------------------------------
// T[v] = LayerNorm(vq_dict[v]; vq_ln_g, vq_ln_b) @ vq_o_w   (4096 x 64 table)
// ---------------------------------------------------------------------------
__global__ __launch_bounds__(64) void build_T_kernel(
    const float* __restrict__ dict, const float* __restrict__ g, const float* __restrict__ bt,
    const float* __restrict__ ow, float* __restrict__ T) {
  int v = blockIdx.x, c = threadIdx.x;
  __shared__ float red[64];
  __shared__ float e_sh[64];
  float z = dict[v*DKV + c];
  red[c] = z; __syncthreads();
  for (int s = 32; s > 0; s >>= 1) { if (c < s) red[c] += red[c+s]; __syncthreads(); }
  float m = red[0] * (1.0f/64.0f); __syncthreads();
  float d = z - m;
  red[c] = d*d; __syncthreads();
  for (int s = 32; s > 0; s >>= 1) { if (c < s) red[c] += red[c+s]; __syncthreads(); }
  float inv = rsqrtf(red[0] * (1.0f/64.0f) + 1e-6f); __syncthreads();
  e_sh[c] = d * inv * g[c] + bt[c];
  __syncthreads();
  float acc = 0.f;
  for (int dd = 0; dd < DKV; ++dd) acc += e_sh[dd] * ow[dd*DKV + c];
  T[v*DKV + c] = acc;
}

// ---------------------------------------------------------------------------
// Heavy kernel: per (b,h), 16x16 (q,k) tile per block; scan vocab in LDS
// chunks of 256 tracking max/argmax of A[q,v]+B[k,v]; then gather T[idx],
// scale by sim, and accumulate the k-mean into accum[b,q,h,:].
// ---------------------------------------------------------------------------
#define VC   256
#define VPAD 264   // 264 % 64 == 8 -> adjacent rows land in different banks

__global__ __launch_bounds__(256) void argmax_kernel(
    const float* __restrict__ Amat, const float* __restrict__ Bmat,
    const float* __restrict__ qn2, const float* __restrict__ kn2,
    const float* __restrict__ G,   const float* __restrict__ T,
    float* __restrict__ accum, float* __restrict__ out_ids, float* __restrict__ out_sc) {
  __shared__ float sA[16][VPAD];
  __shared__ float sB[16][VPAD];
  __shared__ float sSim[16][16];
  __shared__ int   sIdx[16][16];

  int t  = threadIdx.x;
  int bh = blockIdx.z;
  int b  = bh >> 3, h = bh & 7;
  int qt = blockIdx.y * 16, kt = blockIdx.x * 16;
  int qi = t >> 4, ki = t & 15;

  // row bases into A/B (rows are (b,*,h) head-vectors; row stride = 8*4096)
  size_t baseA = ((size_t)((b*QLEN + qt)*NHEADS + h)) * VOCAB;
  size_t baseB = ((size_t)((b*KLEN + kt)*NHEADS + h)) * VOCAB;

  float best = -3.0e38f;
  int bidx = 0;
  for (int c0 = 0; c0 < VOCAB; c0 += VC) {
    // cooperative coalesced load: row i, col t
    for (int i = 0; i < 16; ++i) {
      sA[i][t] = Amat[baseA + (size_t)i * (NHEADS*VOCAB) + c0 + t];
      sB[i][t] = Bmat[baseB + (size_t)i * (NHEADS*VOCAB) + c0 + t];
    }
    __syncthreads();
    const float* aRow = sA[qi];
    const float* bRow = sB[ki];
#pragma unroll 8
    for (int v = 0; v < VC; ++v) {
      float s = aRow[v] + bRow[v];
      if (s > best) { best = s; bidx = c0 + v; }
    }
    __syncthreads();
  }

  int q = qt + qi, k = kt + ki;
  float nrm2 = qn2[(b*QLEN + q)*NHEADS + h]
             + 2.0f * G[((bh)*QLEN + q)*KLEN + k]
             + kn2[(b*KLEN + k)*NHEADS + h];
  float simv = best * rsqrtf(fmaxf(nrm2, 1e-12f));
  int tok = ((b*QLEN + q)*KLEN + k)*NHEADS + h;
  out_ids[tok] = (float)bidx;
  out_sc[tok]  = simv;
  sSim[qi][ki] = simv;
  sIdx[qi][ki] = bidx;
  __syncthreads();

  // reduce over the 16 k's of this tile; atomically add to global k-mean accum
  for (int rep = 0; rep < 4; ++rep) {
    int qr = rep*4 + (t >> 6);
    int c  = t & 63;
    float acc = 0.f;
    for (int kk = 0; kk < 16; ++kk)
      acc += T[sIdx[qr][kk]*DKV + c] * sSim[qr][kk];
    atomicAdd(&accum[((b*QLEN + qt + qr)*NHEADS + h)*DKV + c], acc * (1.0f/64.0f));
  }
}

// ---------------------------------------------------------------------------
// Final LayerNorm + transpose to [b, h, q, d]
// ---------------------------------------------------------------------------
__global__ __launch_bounds__(64) void final_ln_kernel(
    const float* __restrict__ accum, const float* __restrict__ g, const float* __restrict__ bt,
    float* __restrict__ out) {
  int z = blockIdx.x, c = threadIdx.x;
  int b = z >> 9, h = (z >> 6) & 7, q = z & 63;
  float x = accum[((b*QLEN + q)*NHEADS + h)*DKV + c];
  __shared__ float red[64];
  red[c] = x; __syncthreads();
  for (int s = 32; s > 0; s >>= 1) { if (c < s) red[c] += red[c+s]; __syncthreads(); }
  float m = red[0] * (1.0f/64.0f); __syncthreads();
  float d = x - m;
  red[c] = d*d; __syncthreads();
  for (int s = 32; s > 0; s >>= 1) { if (c < s) red[c] += red[c+s]; __syncthreads(); }
  float inv = rsqrtf(red[0] * (1.0f/64.0f) + 1e-6f);
  out[(((b*NHEADS + h)*QLEN + q))*DKV + c] = d * inv * g[c] + bt[c];
}

// ---------------------------------------------------------------------------
extern "C" void kernel_launch(void* const* d_in, const int* in_sizes, int n_in,
                              void* d_out, int out_size, void* d_ws, size_t ws_size,
                              hipStream_t stream) {
  const float* query = (const float*)d_in[0];
  const float* key_  = (const float*)d_in[1];
  const float* qpe   = (const float*)d_in[2];
  const float* kpe   = (const float*)d_in[3];
  const float* wi_w  = (const float*)d_in[4];
  const float* vqd   = (const float*)d_in[5];
  const float* vq_g  = (const float*)d_in[6];
  const float* vq_b  = (const float*)d_in[7];
  const float* vq_ow = (const float*)d_in[8];
  const float* ln_g  = (const float*)d_in[9];
  const float* ln_b  = (const float*)d_in[10];
  (void)in_sizes; (void)n_in; (void)out_size; (void)ws_size;

  float* w     = (float*)d_ws;
  float* qsum  = w;                       // 65536
  float* ksum  = qsum  + 65536;           // 65536
  float* qproj = ksum  + 65536;           // 65536
  float* kproj = qproj + 65536;           // 65536
  float* dn    = kproj + 65536;           // 262144
  float* Amat  = dn    + 262144;          // 4194304
  float* Bmat  = Amat  + 4194304;         // 4194304
  float* qn2   = Bmat  + 4194304;         // 1024
  float* kn2   = qn2   + 1024;            // 1024
  float* G     = kn2   + 1024;            // 65536
  float* T     = G     + 65536;           // 262144
  float* accum = T     + 262144;          // 65536

  float* out_w   = (float*)d_out;         // [2,8,64,64]
  float* out_ids = out_w   + 65536;       // [2,64,64,8]
  float* out_sc  = out_ids + 65536;       // [2,64,64,8]

  // 1) positional add
  addpos_kernel<<<256, 256, 0, stream>>>(query, qpe, key_, kpe, qsum, ksum, BS*QLEN*INNER);

  // 2) q_proj = (q+pe) @ wi_w[:512] ; k_proj = (k+pe) @ wi_w[512:]  (WMMA NN)
  gemm_nn_wmma<<<dim3(INNER/64, MROWS/16, 1), 128, 0, stream>>>(
      qsum, wi_w,               qproj, MROWS, INNER, INNER, INNER, INNER, INNER);
  gemm_nn_wmma<<<dim3(INNER/64, MROWS/16, 1), 128, 0, stream>>>(
      ksum, wi_w + INNER*INNER, kproj, MROWS, INNER, INNER, INNER, INNER, INNER);

  // 3) normalized dictionary
  dictnorm_kernel<<<VOCAB/256, 256, 0, stream>>>(vqd, dn);

  // 4) A = qp_heads @ d_n^T ; B = kp_heads @ d_n^T   (WMMA NT, 1024x4096x64)
  gemm_nt_wmma<<<dim3(VOCAB/64, TOKR/16, 1), 128, 0, stream>>>(
      qproj, dn, Amat, TOKR, VOCAB, DKV, DKV, DKV, VOCAB, 1, 0,0, 0,0, 0,0);
  gemm_nt_wmma<<<dim3(VOCAB/64, TOKR/16, 1), 128, 0, stream>>>(
      kproj, dn, Bmat, TOKR, VOCAB, DKV, DKV, DKV, VOCAB, 1, 0,0, 0,0, 0,0);

  // 5) Gram G[b,h,q,k] = qp_head . kp_head  (batched WMMA NT, 16 x 64x64x64)
  gemm_nt_wmma<<<dim3(1, 4, BS*NHEADS), 128, 0, stream>>>(
      qproj, kproj, G, QLEN, KLEN, DKV, INNER, INNER, KLEN,
      NHEADS, QLEN*INNER, DKV, KLEN*INNER, DKV, NHEADS*QLEN*KLEN, QLEN*KLEN);

  // 6) per-head squared norms
  rownorm2_kernel<<<TOKR/256, 256, 0, stream>>>(qproj, kproj, qn2, kn2);

  // 7) VQ output table T[v] = LN(dict[v]) @ vq_o_w
  build_T_kernel<<<VOCAB, 64, 0, stream>>>(vqd, vq_g, vq_b, vq_ow, T);

  // 8) zero the k-mean accumulator
  zero_kernel<<<256, 256, 0, stream>>>(accum, BS*QLEN*NHEADS*DKV);

  // 9) vocab scan: argmax/max of A+B, sim normalization, gather T, k-mean
  argmax_kernel<<<dim3(KLEN/16, QLEN/16, BS*NHEADS), 256, 0, stream>>>(
      Amat, Bmat, qn2, kn2, G, T, accum, out_ids, out_sc);

  // 10) final LayerNorm + transpose to [b,h,q,d]
  final_ln_kernel<<<BS*NHEADS*QLEN, 64, 0, stream>>>(accum, ln_g, ln_b, out_w);
}